// ContactDiffusion_5634997092631
// MI455X (gfx1250) — compile-verified
//
#include <hip/hip_runtime.h>

#define NN   6144
#define DLAT 512

typedef __attribute__((ext_vector_type(2))) float v2f;
typedef __attribute__((ext_vector_type(8))) float v8f;

// ---------------------------------------------------------------------------
// Pass 1: squared norms of coords rows  -> ws
// ---------------------------------------------------------------------------
__global__ void k_sq(const float* __restrict__ coords, float* __restrict__ sq) {
    int i = blockIdx.x * 256 + threadIdx.x;
    if (i < NN) {
        float x = coords[3 * i + 0];
        float y = coords[3 * i + 1];
        float z = coords[3 * i + 2];
        sq[i] = x * x + y * y + z * z;
    }
}

// ---------------------------------------------------------------------------
// Pass 2: one workgroup per row i.  Compute unnormalized kernel row into LDS,
// reduce the row sum, write the normalized row to global (single HBM write of K).
// ---------------------------------------------------------------------------
__global__ void __launch_bounds__(256) k_rows(const float* __restrict__ coords,
                                              const float* __restrict__ alpha,
                                              const float* __restrict__ sq,
                                              float* __restrict__ Kout) {
    __shared__ float vals[NN];      // 24 KB of the 320 KB/WGP LDS
    __shared__ float ssum;

    const int i   = blockIdx.x;
    const int tid = threadIdx.x;

    const float ci0 = coords[3 * i + 0];
    const float ci1 = coords[3 * i + 1];
    const float ci2 = coords[3 * i + 2];
    const float sqi = sq[i];
    const float ai  = alpha[i];

    if (tid == 0) ssum = 0.0f;
    __syncthreads();

    float local = 0.0f;
    for (int j = tid; j < NN; j += 256) {
        float cx = coords[3 * j + 0];
        float cy = coords[3 * j + 1];
        float cz = coords[3 * j + 2];
        float d2 = sqi + sq[j] - 2.0f * (ci0 * cx + ci1 * cy + ci2 * cz);
        float D  = sqrtf(fmaxf(d2, 1e-12f)) + 1e-6f;
        float a  = 0.5f * (ai + alpha[j]);
        // D^-a = 2^(-a*log2 D);  exp(-D/12) = 2^(-D/(12*ln2))
        float p  = __builtin_amdgcn_exp2f(-a * __builtin_amdgcn_logf(D));
        p        = fminf(fmaxf(p, 1e-8f), 1000.0f);
        float v  = p * __builtin_amdgcn_exp2f(-0.1202245866f * D);
        if (j == i) v = 0.0f;   // zero diagonal
        vals[j] = v;
        local  += v;
    }

    // wave32 butterfly reduce, then one LDS atomic per wave
    for (int m = 16; m >= 1; m >>= 1) local += __shfl_xor(local, m, 32);
    if ((tid & 31) == 0) atomicAdd(&ssum, local);
    __syncthreads();

    const float inv = 1.0f / (ssum + 1e-8f);
    float* row = Kout + (size_t)i * NN;
    for (int j = tid; j < NN; j += 256) row[j] = vals[j] * inv;
}

// ---------------------------------------------------------------------------
// Transpose latent [6144x512] -> LT [512x6144] so B fragments become single
// aligned b64 loads (elements k,k+1 contiguous per output column).
// ---------------------------------------------------------------------------
__global__ void __launch_bounds__(256) k_tr(const float* __restrict__ L,
                                            float* __restrict__ LT) {
    __shared__ float tile[32][33];
    const int n0 = blockIdx.x * 32;   // latent column block
    const int k0 = blockIdx.y * 32;   // latent row block
    const int tx = threadIdx.x;       // 0..31
    const int ty = threadIdx.y;       // 0..7
#pragma unroll
    for (int i = 0; i < 4; ++i)
        tile[ty + 8 * i][tx] = L[(size_t)(k0 + ty + 8 * i) * DLAT + n0 + tx];
    __syncthreads();
#pragma unroll
    for (int i = 0; i < 4; ++i)
        LT[(size_t)(n0 + ty + 8 * i) * NN + k0 + tx] = tile[tx][ty + 8 * i];
}

// ---------------------------------------------------------------------------
// Pass 3 (fast path): out = K @ latent with transposed-B operand.
// Wave = 32 K-rows (2 A frags) x 64 cols (4 B frags) = 8 accumulators, so the
// inner step is 8 v_wmma_f32_16x16x4_f32 per 6 b64 loads.  WG = 8 waves covers
// all 512 columns; K streamed from HBM exactly once.
// ---------------------------------------------------------------------------
__global__ void __launch_bounds__(256) k_gemm_t(const float* __restrict__ Kn,
                                                const float* __restrict__ LT,
                                                float* __restrict__ out) {
    const int tid  = threadIdx.x;
    const int wave = tid >> 5;
    const int lane = tid & 31;
    const int half = lane >> 4;   // selects K-pair / M-offset
    const int col  = lane & 15;

    const int m0 = blockIdx.x * 32;
    const int n0 = wave * 64;

    v8f acc[2][4] = {};

    const float* A0 = Kn + (size_t)(m0 + col) * NN + 2 * half;        // rows m0..m0+15
    const float* A1 = A0 + (size_t)16 * NN;                           // rows m0+16..m0+31
    const float* B0 = LT + (size_t)(n0 + col) * NN + 2 * half;        // cols n0.. (k-contiguous)

    for (int kk = 0; kk < NN; kk += 4) {
        v2f a0 = *(const v2f*)(A0 + kk);
        v2f a1 = *(const v2f*)(A1 + kk);
#pragma unroll
        for (int t = 0; t < 4; ++t) {
            v2f b = *(const v2f*)(B0 + (size_t)(16 * t) * NN + kk);
            acc[0][t] = __builtin_amdgcn_wmma_f32_16x16x4_f32(false, a0, false, b, (short)0, acc[0][t], false, false);
            acc[1][t] = __builtin_amdgcn_wmma_f32_16x16x4_f32(false, a1, false, b, (short)0, acc[1][t], false, false);
        }
    }

#pragma unroll
    for (int p = 0; p < 2; ++p)
#pragma unroll
        for (int t = 0; t < 4; ++t)
#pragma unroll
            for (int v = 0; v < 8; ++v)
                out[(size_t)(m0 + 16 * p + v + 8 * half) * DLAT + n0 + 16 * t + col] = acc[p][t][v];
}

// ---------------------------------------------------------------------------
// Pass 3 (fallback, no scratch for LT): row-major B, pair b32 loads.
// ---------------------------------------------------------------------------
__global__ void __launch_bounds__(256) k_gemm(const float* __restrict__ Kn,
                                              const float* __restrict__ L,
                                              float* __restrict__ out) {
    const int tid  = threadIdx.x;
    const int wave = tid >> 5;
    const int lane = tid & 31;
    const int half = lane >> 4;
    const int col  = lane & 15;

    const int m0 = blockIdx.x * 16;
    const int n0 = wave * 64;

    v8f acc0 = {}, acc1 = {}, acc2 = {}, acc3 = {};

    const float* Arow  = Kn + (size_t)(m0 + col) * NN + 2 * half;
    const float* Bbase = L + (size_t)(2 * half) * DLAT + n0 + col;

    for (int kk = 0; kk < NN; kk += 4) {
        v2f a = *(const v2f*)(Arow + kk);
        const float* Bp = Bbase + (size_t)kk * DLAT;
        v2f b0, b1, b2, b3;
        b0.x = Bp[0];   b0.y = Bp[DLAT + 0];
        b1.x = Bp[16];  b1.y = Bp[DLAT + 16];
        b2.x = Bp[32];  b2.y = Bp[DLAT + 32];
        b3.x = Bp[48];  b3.y = Bp[DLAT + 48];
        acc0 = __builtin_amdgcn_wmma_f32_16x16x4_f32(false, a, false, b0, (short)0, acc0, false, false);
        acc1 = __builtin_amdgcn_wmma_f32_16x16x4_f32(false, a, false, b1, (short)0, acc1, false, false);
        acc2 = __builtin_amdgcn_wmma_f32_16x16x4_f32(false, a, false, b2, (short)0, acc2, false, false);
        acc3 = __builtin_amdgcn_wmma_f32_16x16x4_f32(false, a, false, b3, (short)0, acc3, false, false);
    }

#pragma unroll
    for (int v = 0; v < 8; ++v) {
        size_t r = (size_t)(m0 + v + 8 * half) * DLAT + n0 + col;
        out[r + 0]  = acc0[v];
        out[r + 16] = acc1[v];
        out[r + 32] = acc2[v];
        out[r + 48] = acc3[v];
    }
}

// ---------------------------------------------------------------------------
extern "C" void kernel_launch(void* const* d_in, const int* in_sizes, int n_in,
                              void* d_out, int out_size, void* d_ws, size_t ws_size,
                              hipStream_t stream) {
    const float* latent = (const float*)d_in[0];   // [N, 512]
    const float* coords = (const float*)d_in[1];   // [N, 3]
    const float* alpha  = (const float*)d_in[2];   // [N]

    float* out  = (float*)d_out;                   // [N, 512]  (tuple slot 0)
    float* Kmat = out + (size_t)NN * DLAT;         // [N, N]    (tuple slot 1)

    float*       sq    = (float*)d_ws;                    // [N] scratch
    const size_t ltOff = 8192;                            // floats (32 KB, keeps LT aligned)
    const size_t need  = (ltOff + (size_t)DLAT * NN) * sizeof(float);

    k_sq  <<<(NN + 255) / 256, 256, 0, stream>>>(coords, sq);
    k_rows<<<NN,              256, 0, stream>>>(coords, alpha, sq, Kmat);

    if (ws_size >= need) {
        float* LT = (float*)d_ws + ltOff;          // [512, 6144]
        k_tr    <<<dim3(DLAT / 32, NN / 32), dim3(32, 8), 0, stream>>>(latent, LT);
        k_gemm_t<<<NN / 32, 256, 0, stream>>>(Kmat, LT, out);
    } else {
        k_gemm  <<<NN / 16, 256, 0, stream>>>(Kmat, latent, out);
    }
}